// LSTMcell_10720238370825
// MI455X (gfx1250) — compile-verified
//
#include <hip/hip_runtime.h>

typedef float v2f __attribute__((ext_vector_type(2)));
typedef float v8f __attribute__((ext_vector_type(8)));

#define BATCH   16
#define NSTOCK  1000
#define INDIM   128
#define OUTDIM  128
#define KDIM    256          // IN + OUT
#define LDSTR   260          // padded row stride (dwords) -> conflict-free A reads

__device__ __forceinline__ float sigmoid_f(float x) {
    x = fminf(fmaxf(x, -30.0f), 30.0f);
    return 1.0f / (1.0f + __expf(-x));
}
__device__ __forceinline__ float tanh_f(float x) {
    x = fminf(fmaxf(x, -15.0f), 15.0f);
    float e = __expf(2.0f * x);
    return (e - 1.0f) / (e + 1.0f);
}

__global__ __launch_bounds__(256) void lstm_wmma_f32_kernel(
    const float* __restrict__ xt,     // [B, N, IN]
    const float* __restrict__ hidden, // [1, N, OUT]
    const float* __restrict__ ct_1,   // [1, N, OUT]
    const float* __restrict__ W_i, const float* __restrict__ b_i,
    const float* __restrict__ W_o, const float* __restrict__ b_o,
    const float* __restrict__ W_f, const float* __restrict__ b_f,
    const float* __restrict__ W_c, const float* __restrict__ b_c,
    float* __restrict__ ht_out,       // [B, N, OUT]
    float* __restrict__ ct_out)       // [B, N, OUT]
{
    __shared__ float sA[BATCH * LDSTR];   // xh = [xt | h], 16 x 256 (padded)

    const int n   = blockIdx.x;
    const int tid = threadIdx.x;

    // ---- stage xh into LDS (coalesced along K) ----
    for (int i = tid; i < BATCH * KDIM; i += 256) {
        const int m = i >> 8;          // batch row
        const int c = i & 255;         // K column
        float v = (c < INDIM)
                    ? xt[(size_t)m * (NSTOCK * INDIM) + (size_t)n * INDIM + c]
                    : hidden[(size_t)n * OUTDIM + (c - INDIM)];
        sA[m * LDSTR + c] = v;
    }
    __syncthreads();

    const int lane = tid & 31;
    const int wave = tid >> 5;
    const int ncol = lane & 15;           // N within tile / M for A reads
    const int o    = wave * 16 + ncol;    // global output column (0..127)
    const int krow = (lane >> 4) * 2;     // K sub-row: lanes 16-31 offset by 2

    const size_t wbase = (size_t)n * KDIM * OUTDIM;
    const float* Wi = W_i + wbase;
    const float* Wo = W_o + wbase;
    const float* Wf = W_f + wbase;
    const float* Wc = W_c + wbase;

    v8f acc_i = {}, acc_o = {}, acc_f = {}, acc_c = {};

    #pragma unroll 4
    for (int kk = 0; kk < KDIM; kk += 4) {
        const int kb = kk + krow;
        // A fragment (16x4 f32): lane holds xh[M = lane&15][kb], xh[..][kb+1]
        v2f a = *(const v2f*)&sA[ncol * LDSTR + kb];

        // B fragments (4x16 f32): lane holds W[kb][o], W[kb+1][o]
        const size_t off = (size_t)kb * OUTDIM + o;
        v2f bi, bo, bf, bc;
        bi.x = __builtin_nontemporal_load(&Wi[off]);
        bi.y = __builtin_nontemporal_load(&Wi[off + OUTDIM]);
        bo.x = __builtin_nontemporal_load(&Wo[off]);
        bo.y = __builtin_nontemporal_load(&Wo[off + OUTDIM]);
        bf.x = __builtin_nontemporal_load(&Wf[off]);
        bf.y = __builtin_nontemporal_load(&Wf[off + OUTDIM]);
        bc.x = __builtin_nontemporal_load(&Wc[off]);
        bc.y = __builtin_nontemporal_load(&Wc[off + OUTDIM]);

        acc_i = __builtin_amdgcn_wmma_f32_16x16x4_f32(false, a, false, bi,
                                                      (short)0, acc_i, false, false);
        acc_o = __builtin_amdgcn_wmma_f32_16x16x4_f32(false, a, false, bo,
                                                      (short)0, acc_o, false, false);
        acc_f = __builtin_amdgcn_wmma_f32_16x16x4_f32(false, a, false, bf,
                                                      (short)0, acc_f, false, false);
        acc_c = __builtin_amdgcn_wmma_f32_16x16x4_f32(false, a, false, bc,
                                                      (short)0, acc_c, false, false);
    }

    // ---- fused epilogue: bias + activations + cell update (all in registers) ----
    const size_t no = (size_t)n * OUTDIM + o;
    const float bias_i = b_i[no];
    const float bias_o = b_o[no];
    const float bias_f = b_f[no];
    const float bias_c = b_c[no];
    const float c_prev = ct_1[no];        // broadcast across batch
    const int   mhi    = (lane >> 4) * 8; // C/D layout: VGPR r -> M = r + 8*(lane>=16)

    #pragma unroll
    for (int r = 0; r < 8; ++r) {
        const int m = r + mhi;
        const float it = sigmoid_f(acc_i[r] + bias_i);
        const float ot = sigmoid_f(acc_o[r] + bias_o);
        const float ft = sigmoid_f(acc_f[r] + bias_f);
        const float ch = tanh_f(acc_c[r] + bias_c);
        const float ct = ft * c_prev + it * ch;
        const float ht = ot * tanh_f(ct);
        const size_t oidx = (size_t)m * (NSTOCK * OUTDIM) + no;
        ht_out[oidx] = ht;
        ct_out[oidx] = ct;
    }
}

extern "C" void kernel_launch(void* const* d_in, const int* in_sizes, int n_in,
                              void* d_out, int out_size, void* d_ws, size_t ws_size,
                              hipStream_t stream) {
    const float* xt     = (const float*)d_in[0];
    const float* hidden = (const float*)d_in[1];
    const float* ct_1   = (const float*)d_in[2];
    const float* W_i    = (const float*)d_in[3];
    const float* b_i    = (const float*)d_in[4];
    const float* W_o    = (const float*)d_in[5];
    const float* b_o    = (const float*)d_in[6];
    const float* W_f    = (const float*)d_in[7];
    const float* b_f    = (const float*)d_in[8];
    const float* W_c    = (const float*)d_in[9];
    const float* b_c    = (const float*)d_in[10];

    float* ht = (float*)d_out;
    float* ct = ht + (size_t)BATCH * NSTOCK * OUTDIM;

    lstm_wmma_f32_kernel<<<NSTOCK, 256, 0, stream>>>(
        xt, hidden, ct_1, W_i, b_i, W_o, b_o, W_f, b_f, W_c, b_c, ht, ct);
}